// NovelViewHomography_52965536694458
// MI455X (gfx1250) — compile-verified
//
#include <hip/hip_runtime.h>
#include <hip/hip_bf16.h>
#include <math.h>

typedef float v2f __attribute__((ext_vector_type(2)));
typedef float v8f __attribute__((ext_vector_type(8)));

#define B_ 8
#define C_ 3
#define H_ 1024
#define W_ 2048
#define M_ 32
#define h_ 128
#define w_ 256
#define HW_ (h_*w_)          /* 32768 */
#define NSPLIT 16
#define KRANGE (HW_/NSPLIT)  /* 2048 */
#define TK 128

/* workspace layout (float offsets) */
#define OFF_DEPTH   0                                   /* B*HW          = 262144 */
#define OFF_NORMAL  (OFF_DEPTH + B_*HW_)                /* B*3*HW        = 786432 */
#define OFF_IMGP    (OFF_NORMAL + B_*3*HW_)             /* B*HW*3 (bhwc) = 786432 */
#define OFF_ROWMAX  (OFF_IMGP + B_*HW_*3)               /* B*M           = 256    */
#define OFF_PART    (OFF_ROWMAX + B_*M_)                /* B*2*NSPLIT*256= 65536  */
#define OFF_HROW    (OFF_PART + B_*2*NSPLIT*256)        /* B*M*8         = 2048   */
/* total ~1,902,848 floats (~7.6 MB) */

/* ---------------- 1: disp -> depth ---------------- */
__global__ void k_depth(const float* __restrict__ disp, const float* __restrict__ intrs,
                        const float* __restrict__ baseline, float* __restrict__ ws) {
    int idx = blockIdx.x * 256 + threadIdx.x;      /* b*HW + p */
    int b = idx / HW_;
    float focal = (intrs[b*9 + 0] + intrs[b*9 + 4]) * 0.5f;
    ws[OFF_DEPTH + idx] = baseline[b] * focal / (disp[idx] * 2048.0f);
}

/* ---------------- 2: depth -> normals (channel-major (B,3,h,w)) ---------------- */
__global__ void k_normal(float* __restrict__ ws) {
    int idx = blockIdx.x * 256 + threadIdx.x;
    int b = idx / HW_;
    int p = idx - b * HW_;
    int y = p >> 8, x = p & (w_-1);
    const float* dep = ws + OFF_DEPTH + (size_t)b * HW_;
    float dl = (x > 0)     ? dep[p-1]  : 0.0f;
    float dr = (x < w_-1)  ? dep[p+1]  : 0.0f;
    float du = (y > 0)     ? dep[p-w_] : 0.0f;
    float dd = (y < h_-1)  ? dep[p+w_] : 0.0f;
    float gx = 0.5f * (dr - dl);
    float gy = 0.5f * (dd - du);
    float inv = 1.0f / sqrtf(gx*gx + gy*gy + 1.0f);
    float* nrm = ws + OFF_NORMAL + (size_t)b * 3 * HW_;
    nrm[0*HW_ + p] = -gx * inv;
    nrm[1*HW_ + p] = -gy * inv;
    nrm[2*HW_ + p] =  inv;
}

/* ---------------- 3: per-(b,m) row max over h*w ---------------- */
__global__ void k_rowmax(const float* __restrict__ pm, float* __restrict__ ws) {
    int row = blockIdx.x;                           /* b*M + m */
    const float* src = pm + (size_t)row * HW_;
    float mx = -INFINITY;
    for (int i = threadIdx.x; i < HW_; i += 256) mx = fmaxf(mx, src[i]);
    __shared__ float red[256];
    red[threadIdx.x] = mx;
    __syncthreads();
    for (int s = 128; s > 0; s >>= 1) {
        if (threadIdx.x < s) red[threadIdx.x] = fmaxf(red[threadIdx.x], red[threadIdx.x + s]);
        __syncthreads();
    }
    if (threadIdx.x == 0) ws[OFF_ROWMAX + row] = red[0];
}

/* ---------------- 4: WMMA weighted reduction ----------------
 * Per batch b: D(32x5) = exp(mask - rowmax) (32 x 32768)  X  F (32768 x 5)
 * F columns: [depth, n(3k), n(3k+1), n(3k+2), 1.0]  (col 4 of ones -> softmax Z)
 * V_WMMA_F32_16X16X4_F32 per K-chunk of 4; chained accumulator.
 * Grid: b(8) x mtile(2) x ksplit(16) = 256 blocks of 256 threads (8 waves).
 */
__global__ void k_wmma_reduce(const float* __restrict__ pm, float* __restrict__ ws) {
    int blk   = blockIdx.x;
    int split = blk & (NSPLIT - 1);
    int mtile = (blk >> 4) & 1;
    int b     = blk >> 5;
    int tid   = threadIdx.x;
    int lane  = tid & 31, wave = tid >> 5;
    int mbase = mtile * 16;
    int kbase0 = split * KRANGE;

    __shared__ float maskT[16 * 132];   /* padded stride: banks differ across rows */
    __shared__ float dbuf[TK];
    __shared__ float nbuf[3 * TK];
    __shared__ float rmax[16];
    __shared__ float red[8 * 256];

    if (tid < 16) rmax[tid] = ws[OFF_ROWMAX + b * M_ + mbase + tid];
    __syncthreads();

    v8f acc = {0.f, 0.f, 0.f, 0.f, 0.f, 0.f, 0.f, 0.f};
    const float* dep = ws + OFF_DEPTH  + (size_t)b * HW_;
    const float* nrm = ws + OFF_NORMAL + (size_t)b * 3 * HW_;
    const float* pmb = pm + ((size_t)b * M_ + mbase) * HW_;

    for (int kb = kbase0; kb < kbase0 + KRANGE; kb += TK) {
        for (int i = tid; i < 16 * TK; i += 256) {
            int r = i >> 7, c = i & (TK - 1);
            maskT[r * 132 + c] = expf(pmb[(size_t)r * HW_ + kb + c] - rmax[r]);
        }
        if (tid < TK) dbuf[tid] = dep[kb + tid];
        for (int i = tid; i < 3 * TK; i += 256) nbuf[i] = nrm[(size_t)3 * kb + i];
        __syncthreads();

        /* A(16x4): lanes0-15 row=lane, VGPR0/1 = K {0,2}/{1,3}.  B(4x16) mirrored. */
        int mrow  = lane & 15;
        int khalf = (lane >> 4) << 1;   /* 0 for lanes 0-15, 2 for lanes 16-31 */
        int n     = lane & 15;          /* B/D column */
        for (int kc4 = 0; kc4 < 16; kc4 += 4) {
            int kc = wave * 16 + kc4 + khalf;
            v2f a, bb;
            a[0] = maskT[mrow * 132 + kc];
            a[1] = maskT[mrow * 132 + kc + 1];
            bb[0] = (n == 0) ? dbuf[kc]     : (n <= 3) ? nbuf[3*kc     + (n-1)] : (n == 4) ? 1.0f : 0.0f;
            bb[1] = (n == 0) ? dbuf[kc + 1] : (n <= 3) ? nbuf[3*(kc+1) + (n-1)] : (n == 4) ? 1.0f : 0.0f;
            acc = __builtin_amdgcn_wmma_f32_16x16x4_f32(
                false, a, false, bb, (short)0, acc, false, false);
        }
        __syncthreads();
    }

    /* cross-wave reduce in (vgpr,lane) coordinates, fixed order -> deterministic */
#pragma unroll
    for (int v = 0; v < 8; ++v) red[wave * 256 + v * 32 + lane] = acc[v];
    __syncthreads();
    float s = 0.f;
#pragma unroll
    for (int wv = 0; wv < 8; ++wv) s += red[wv * 256 + tid];
    ws[OFF_PART + ((size_t)(b * 2 + mtile) * NSPLIT + split) * 256 + tid] = s;
}

/* ---------------- 5: reduce splits, build homography rows ---------------- */
__global__ void k_finalize(const float* __restrict__ intrs, const float* __restrict__ baseline,
                           float* __restrict__ ws) {
    int t = threadIdx.x;                 /* one thread per (b,m): 256 */
    int b = t >> 5, m = t & 31;
    int mtile = m >> 4, mrow = m & 15;
    const float* part = ws + OFF_PART + (size_t)(b * 2 + mtile) * NSPLIT * 256;
    /* D layout: element (m,n) lives at vgpr = m&7, lane = n + 16*(m>=8) */
    int vv = mrow & 7;
    int lh = (mrow >> 3) << 4;
    float val[5];
    for (int n = 0; n < 5; ++n) {
        int idx = vv * 32 + lh + n;
        float s = 0.f;
        for (int sp = 0; sp < NSPLIT; ++sp) s += part[sp * 256 + idx];
        val[n] = s;
    }
    float Z  = val[4];
    float pd = val[0] / Z;
    float n0 = val[1] / Z, n1 = val[2] / Z, n2 = val[3] / Z;

    const float* K = intrs + b * 9;
    float a00=K[0],a01=K[1],a02=K[2],a10=K[3],a11=K[4],a12=K[5],a20=K[6],a21=K[7],a22=K[8];
    float det = a00*(a11*a22-a12*a21) - a01*(a10*a22-a12*a20) + a02*(a10*a21-a11*a20);
    float id = 1.0f / det;
    float i00 =  (a11*a22-a12*a21)*id, i01 = -(a01*a22-a02*a21)*id, i02 =  (a01*a12-a02*a11)*id;
    float i10 = -(a10*a22-a12*a20)*id, i11 =  (a00*a22-a02*a20)*id, i12 = -(a00*a12-a02*a10)*id;
    float i20 =  (a10*a21-a11*a20)*id, i21 = -(a00*a21-a01*a20)*id, i22 =  (a00*a11-a01*a10)*id;

    float bl = baseline[b];
    /* Hcal = I, row2 = [bl*n0/d, bl*n1/d, 1 + bl*n2/d] */
    float r20 = bl*n0/pd, r21 = bl*n1/pd, r22 = 1.0f + bl*n2/pd;
    /* T = Hcal * Kinv */
    float t00=i00,t01=i01,t02=i02, t10=i10,t11=i11,t12=i12;
    float t20 = r20*i00 + r21*i10 + r22*i20;
    float t21 = r20*i01 + r21*i11 + r22*i21;
    float t22 = r20*i02 + r21*i12 + r22*i22;
    /* rows 0,1 of K*T */
    float* o = ws + OFF_HROW + (size_t)(b * M_ + m) * 8;
    o[0] = a00*t00 + a01*t10 + a02*t20;
    o[1] = a00*t01 + a01*t11 + a02*t21;
    o[2] = a00*t02 + a01*t12 + a02*t22;
    o[3] = a10*t00 + a11*t10 + a12*t20;
    o[4] = a10*t01 + a11*t11 + a12*t21;
    o[5] = a10*t02 + a11*t12 + a12*t22;
    o[6] = 0.f; o[7] = 0.f;
}

/* ---------------- 6: 8x8 average pool, store (B,h,w,C) ---------------- */
__global__ void k_avgpool(const float* __restrict__ img, float* __restrict__ ws) {
    int idx = blockIdx.x * 256 + threadIdx.x;       /* ((b*C+c)*h + yo)*w + xo */
    int xo = idx & (w_ - 1);
    int t  = idx >> 8;
    int yo = t & (h_ - 1);
    t >>= 7;
    int c = t % C_;
    int b = t / C_;
    const float* src = img + (((size_t)(b * C_ + c) * H_ + yo * 8) * W_ + xo * 8);
    float s = 0.f;
#pragma unroll
    for (int r = 0; r < 8; ++r) {
        const float4* p4 = (const float4*)(src + (size_t)r * W_);
        float4 u = p4[0], v = p4[1];
        s += u.x + u.y + u.z + u.w + v.x + v.y + v.z + v.w;
    }
    ws[OFF_IMGP + ((size_t)(b * h_ + yo) * w_ + xo) * 3 + c] = s * (1.0f / 64.0f);
}

/* ---------------- 7: per-pixel softmax over M, warp + blend ---------------- */
__global__ void k_warp(const float* __restrict__ pm, const float* __restrict__ ws,
                       float* __restrict__ out) {
    int b   = blockIdx.x >> 7;
    int pix = ((blockIdx.x & 127) << 8) + threadIdx.x;
    int y = pix >> 8, x = pix & (w_ - 1);

    __shared__ float Hs[256];   /* 32 planes x 8 floats */
    Hs[threadIdx.x] = ws[OFF_HROW + (size_t)b * 256 + threadIdx.x];
    __syncthreads();

    const float* pmb = pm + (size_t)b * M_ * HW_ + pix;
    float mv[M_];
    float vmax = -INFINITY;
#pragma unroll
    for (int m = 0; m < M_; ++m) { mv[m] = pmb[(size_t)m * HW_]; vmax = fmaxf(vmax, mv[m]); }
    float S = 0.f;
#pragma unroll
    for (int m = 0; m < M_; ++m) { mv[m] = expf(mv[m] - vmax); S += mv[m]; }

    float gx = x * (2.0f / (w_ - 1)) - 1.0f;
    float gy = y * (2.0f / (h_ - 1)) - 1.0f;
    const float* imgT = ws + OFF_IMGP + (size_t)b * HW_ * 3;
    float acc0 = 0.f, acc1 = 0.f, acc2 = 0.f;

    auto sample = [&](float xf, float yf, float wgt) {
        bool valid = (xf >= 0.f) && (xf <= (float)(w_ - 1)) &&
                     (yf >= 0.f) && (yf <= (float)(h_ - 1));
        if (valid) {
            int xi = (int)fminf(fmaxf(xf, 0.f), (float)(w_ - 1));
            int yi = (int)fminf(fmaxf(yf, 0.f), (float)(h_ - 1));
            const float* p = imgT + ((size_t)yi * w_ + xi) * 3;
            acc0 += p[0] * wgt; acc1 += p[1] * wgt; acc2 += p[2] * wgt;
        }
    };

    for (int m = 0; m < M_; ++m) {
        const float* Hr = Hs + m * 8;
        float u = Hr[0] * gx + Hr[1] * gy + Hr[2];
        float v = Hr[3] * gx + Hr[4] * gy + Hr[5];
        float ix = (u + 1.0f) * ((w_ - 1) * 0.5f);
        float iy = (v + 1.0f) * ((h_ - 1) * 0.5f);
        float x0 = floorf(ix), y0 = floorf(iy);
        float wx1 = ix - x0, wy1 = iy - y0;
        float wx0 = 1.0f - wx1, wy0 = 1.0f - wy1;
        float e = mv[m];
        sample(x0,        y0,        wx0 * wy0 * e);
        sample(x0 + 1.0f, y0,        wx1 * wy0 * e);
        sample(x0,        y0 + 1.0f, wx0 * wy1 * e);
        sample(x0 + 1.0f, y0 + 1.0f, wx1 * wy1 * e);
    }
    float invS = 1.0f / S;
    size_t ob = (size_t)b * C_ * HW_ + pix;
    out[ob]            = acc0 * invS;
    out[ob +     HW_]  = acc1 * invS;
    out[ob + 2 * HW_]  = acc2 * invS;
}

extern "C" void kernel_launch(void* const* d_in, const int* in_sizes, int n_in,
                              void* d_out, int out_size, void* d_ws, size_t ws_size,
                              hipStream_t stream) {
    const float* img      = (const float*)d_in[0];
    const float* pm       = (const float*)d_in[1];
    const float* disp     = (const float*)d_in[2];
    const float* intrs    = (const float*)d_in[3];
    const float* baseline = (const float*)d_in[4];
    float* ws  = (float*)d_ws;
    float* out = (float*)d_out;

    k_depth      <<<B_*HW_/256,    256, 0, stream>>>(disp, intrs, baseline, ws);
    k_normal     <<<B_*HW_/256,    256, 0, stream>>>(ws);
    k_rowmax     <<<B_*M_,         256, 0, stream>>>(pm, ws);
    k_wmma_reduce<<<B_*2*NSPLIT,   256, 0, stream>>>(pm, ws);
    k_finalize   <<<1,             256, 0, stream>>>(intrs, baseline, ws);
    k_avgpool    <<<B_*C_*HW_/256, 256, 0, stream>>>(img, ws);
    k_warp       <<<B_*HW_/256,    256, 0, stream>>>(pm, ws, out);
}